// SeparateTSQNodeBlock_84198538871459
// MI455X (gfx1250) — compile-verified
//
#include <hip/hip_runtime.h>

typedef float v2f __attribute__((ext_vector_type(2)));
typedef float v8f __attribute__((ext_vector_type(8)));

#define DIM 104
#define TILE 16
#define LDS_STRIDE 108           // floats; 432B rows: 16B aligned, conflict-free over 16 lanes
#define WPB 4                    // waves per block
#define THREADS (WPB * 32)
#define NFRAG 352                // B fragments per weight set (sum over paths of U*V/4)

struct CG { int i, j, k; float v; };

// Real-basis Clebsch-Gordan tensors (unit Frobenius norm), nnz lists sorted by k.
constexpr CG CG_P0[] = {{0,0,0, 1.00000000f}};                                   // 0x0 -> 0
constexpr CG CG_P1[] = {{0,0,0,0.57735027f},{1,0,1,0.57735027f},{2,0,2,0.57735027f}}; // 1x0 -> 1
constexpr CG CG_P2[] = {{0,0,0,0.57735027f},{1,1,0,0.57735027f},{2,2,0,0.57735027f}}; // 1x1 -> 0
constexpr CG CG_P3[] = {                                                          // 1x1 -> 2
  {0,2,0, 0.31622777f},{2,0,0, 0.31622777f},
  {0,1,1, 0.31622777f},{1,0,1, 0.31622777f},
  {0,0,2,-0.18257419f},{1,1,2, 0.36514837f},{2,2,2,-0.18257419f},
  {1,2,3, 0.31622777f},{2,1,3, 0.31622777f},
  {0,0,4,-0.31622777f},{2,2,4, 0.31622777f}};
constexpr CG CG_P4[] = {{0,0,0,0.44721360f},{1,0,1,0.44721360f},{2,0,2,0.44721360f},
                        {3,0,3,0.44721360f},{4,0,4,0.44721360f}};                 // 2x0 -> 2
constexpr CG CG_P5[] = {                                                          // 2x1 -> 1
  {0,2,0, 0.31622777f},{1,1,0, 0.31622777f},{2,0,0,-0.18257419f},{4,0,0,-0.31622777f},
  {1,0,1, 0.31622777f},{2,1,1, 0.36514837f},{3,2,1, 0.31622777f},
  {0,0,2, 0.31622777f},{2,2,2,-0.18257419f},{3,1,2, 0.31622777f},{4,2,2, 0.31622777f}};
constexpr CG CG_P6[] = {{0,0,0,0.44721360f},{1,1,0,0.44721360f},{2,2,0,0.44721360f},
                        {3,3,0,0.44721360f},{4,4,0,0.44721360f}};                 // 2x2 -> 0
constexpr CG CG_P7[] = {                                                          // 2x2 -> 2
  {0,2,0,-0.23904572f},{2,0,0,-0.23904572f},{1,3,0, 0.20701967f},{3,1,0, 0.20701967f},
  {1,2,1, 0.11952286f},{2,1,1, 0.11952286f},{1,4,1,-0.20701967f},{4,1,1,-0.20701967f},
  {0,3,1, 0.20701967f},{3,0,1, 0.20701967f},
  {0,0,2,-0.23904572f},{1,1,2, 0.11952286f},{3,3,2, 0.11952286f},{4,4,2,-0.23904572f},
  {2,2,2, 0.23904572f},
  {3,2,3, 0.11952286f},{2,3,3, 0.11952286f},{3,4,3, 0.20701967f},{4,3,3, 0.20701967f},
  {0,1,3, 0.20701967f},{1,0,3, 0.20701967f},
  {4,2,4,-0.23904572f},{2,4,4,-0.23904572f},{1,1,4,-0.20701967f},{3,3,4, 0.20701967f}};

struct WPtrs { const float* w[16]; };

struct PathMeta { int fragBase, U, V, W; float alpha; };
__constant__ PathMeta PM[8] = {
  {  0, 16, 16, 16, 0.03608439f},
  { 64, 16, 16, 16, 0.07654655f},
  {128, 16, 16, 16, 0.03608439f},
  {192, 16, 16,  8, 0.08068715f},
  {256,  8, 16,  8, 0.11410887f},
  {288,  8, 16, 16, 0.10825318f},
  {320,  8,  8, 16, 0.07216878f},
  {336,  8,  8,  8, 0.16137431f}};

// ---------------------------------------------------------------------------
// Prep: reorder all 16 weight tensors into WMMA B-fragment order in scratch.
// Fragment (s,p,u,cv) = 64 floats at scratch[frag*64]: element [lane*2+r] is
// alpha_p * w[u, cv*4 + 2*(lane>>4) + r, lane&15], pre-zeroed for n >= W.
// Main kernel then reads one coalesced b64 per fragment per lane.
// ---------------------------------------------------------------------------
__global__ void prep_kernel(WPtrs wp, float* __restrict__ scratch) {
  const int frag = blockIdx.x;            // 0 .. 2*NFRAG-1
  const int s    = frag / NFRAG;
  const int f    = frag % NFRAG;
  int p = 0;
#pragma unroll
  for (int q = 1; q < 8; ++q) if (f >= PM[q].fragBase) p = q;
  const PathMeta m = PM[p];
  const int rel = f - m.fragBase;
  const int nCv = m.V / 4;
  const int u = rel / nCv, cv = rel % nCv;

  const int t    = threadIdx.x;           // 64 threads: t = lane*2 + r
  const int r    = t & 1;
  const int lane = t >> 1;
  const int n    = lane & 15;
  const int hi   = lane >> 4;
  const int v    = cv * 4 + 2 * hi + r;

  const float* w = wp.w[s * 8 + p];
  const float val = (n < m.W) ? m.alpha * w[(u * m.V + v) * m.W + n] : 0.0f;
  scratch[(size_t)frag * 64 + t] = val;
}

// ---------------------------------------------------------------------------
// One path: t[node, n, k] += sum_uv Y[node,uv,k] * Bfrag(u,v,n) via
// V_WMMA_F32_16X16X4_F32 over 4-wide uv chunks. A: 16 nodes x 4 uv (built from
// CG outer products on LDS-resident x); B: prebuilt fragment stream.
// ---------------------------------------------------------------------------
template<int U, int V, int D1, int D2, int D3, int OFF1, int OFF2, int ACCB, int NNZ>
__device__ __forceinline__ void tsq_path(const v2f* __restrict__ wfrag,  // path+set base
                                         const float* __restrict__ xrow, // LDS row (lane's node)
                                         int lane, int hi, v8f* acc, const CG (&cg)[NNZ])
{
#pragma unroll 2
  for (int u = 0; u < U; ++u) {
    float x1v[D1];
#pragma unroll
    for (int i = 0; i < D1; ++i) x1v[i] = xrow[OFF1 + u * D1 + i];
#pragma unroll
    for (int cv = 0; cv < V / 4; ++cv) {
      const v2f bf = wfrag[(u * (V / 4) + cv) * 32 + lane];   // coalesced b64, alpha baked in
      const int v0 = cv * 4 + 2 * hi;                          // this lane's two K rows
      float x2a[D2], x2b[D2];
#pragma unroll
      for (int j = 0; j < D2; ++j) {
        x2a[j] = xrow[OFF2 + v0 * D2 + j];
        x2b[j] = xrow[OFF2 + (v0 + 1) * D2 + j];
      }
      float ya[D3], yb[D3];
#pragma unroll
      for (int k = 0; k < D3; ++k) { ya[k] = 0.0f; yb[k] = 0.0f; }
#pragma unroll
      for (int e = 0; e < NNZ; ++e) {                          // CG outer product (constexpr)
        ya[cg[e].k] = __builtin_fmaf(cg[e].v * x1v[cg[e].i], x2a[cg[e].j], ya[cg[e].k]);
        yb[cg[e].k] = __builtin_fmaf(cg[e].v * x1v[cg[e].i], x2b[cg[e].j], yb[cg[e].k]);
      }
#pragma unroll
      for (int k = 0; k < D3; ++k) {
        v2f af; af.x = ya[k]; af.y = yb[k];
        acc[ACCB + k] = __builtin_amdgcn_wmma_f32_16x16x4_f32(
            false, af, false, bf, (short)0, acc[ACCB + k], false, false);
      }
    }
  }
}

__global__ __launch_bounds__(THREADS)
void tsq_kernel(const float* __restrict__ feats, const float* __restrict__ msgs,
                float* __restrict__ out, const float* __restrict__ wscratch, int nNodes)
{
  extern __shared__ float lds[];
  const int wave  = threadIdx.x >> 5;
  const int lane  = threadIdx.x & 31;
  const int tile  = blockIdx.x * WPB + wave;
  const int node0 = tile * TILE;
  if (node0 >= nNodes) return;                        // wave-uniform exit

  float* ldsF = lds + wave * (2 * TILE * LDS_STRIDE);
  float* ldsM = ldsF + TILE * LDS_STRIDE;

  // Stage 16x104 f32 tiles of feats and msgs into LDS (float4 global -> b128 DS).
  for (int idx = lane; idx < TILE * (DIM / 4); idx += 32) {
    const int r = idx / (DIM / 4), c4 = idx % (DIM / 4);
    if (node0 + r < nNodes) {
      const float4 vF = reinterpret_cast<const float4*>(feats + (size_t)(node0 + r) * DIM)[c4];
      const float4 vM = reinterpret_cast<const float4*>(msgs  + (size_t)(node0 + r) * DIM)[c4];
      *reinterpret_cast<float4*>(ldsF + r * LDS_STRIDE + c4 * 4) = vF;
      *reinterpret_cast<float4*>(ldsM + r * LDS_STRIDE + c4 * 4) = vM;
    }
  }
  asm volatile("s_wait_dscnt 0" ::: "memory");        // LDS writes visible to all lanes of wave

  const int n  = lane & 15;                           // node row (A) / output mul (D)
  const int hi = lane >> 4;

  v8f acc[9];                                         // (io0,k0) (io1,k0..2) (io2,k0..4)
#pragma unroll
  for (int a = 0; a < 9; ++a)
#pragma unroll
    for (int r = 0; r < 8; ++r) acc[a][r] = 0.0f;

  const float* rowF = ldsF + n * LDS_STRIDE;
  const float* rowM = ldsM + n * LDS_STRIDE;
  const v2f*   wf0  = reinterpret_cast<const v2f*>(wscratch);

#pragma unroll 1
  for (int s = 0; s < 2; ++s) {                       // feats(ws1) then messages(ws2)
    const float* xr = s ? rowM : rowF;
    const v2f*   wb = wf0 + (size_t)s * NFRAG * 32;
    //        U  V  D1 D2 D3 OFF1 OFF2 ACC
    tsq_path<16,16, 1, 1, 1,  0,   0,  0>(wb +   0 * 32, xr, lane, hi, acc, CG_P0);
    tsq_path<16,16, 3, 1, 3, 16,   0,  1>(wb +  64 * 32, xr, lane, hi, acc, CG_P1);
    tsq_path<16,16, 3, 3, 1, 16,  16,  0>(wb + 128 * 32, xr, lane, hi, acc, CG_P2);
    tsq_path<16,16, 3, 3, 5, 16,  16,  4>(wb + 192 * 32, xr, lane, hi, acc, CG_P3);
    tsq_path< 8,16, 5, 1, 5, 64,   0,  4>(wb + 256 * 32, xr, lane, hi, acc, CG_P4);
    tsq_path< 8,16, 5, 3, 3, 64,  16,  1>(wb + 288 * 32, xr, lane, hi, acc, CG_P5);
    tsq_path< 8, 8, 5, 5, 1, 64,  64,  0>(wb + 320 * 32, xr, lane, hi, acc, CG_P6);
    tsq_path< 8, 8, 5, 5, 5, 64,  64,  4>(wb + 336 * 32, xr, lane, hi, acc, CG_P7);
  }

  // D layout: VGPR r holds node (r + 8*hi), column n. Output col = off + n*d + k.
  float* orow = out + (size_t)(node0 + 8 * hi) * DIM;
#pragma unroll
  for (int r = 0; r < 8; ++r) {
    if (node0 + 8 * hi + r >= nNodes) break;
    orow[r * DIM + n] = acc[0][r];                    // io0: 16x0e @ col 0
#pragma unroll
    for (int k = 0; k < 3; ++k)                       // io1: 16x1o @ col 16
      orow[r * DIM + 16 + n * 3 + k] = acc[1 + k][r];
    if (n < 8) {                                      // io2: 8x2e @ col 64
#pragma unroll
      for (int k = 0; k < 5; ++k)
        orow[r * DIM + 64 + n * 5 + k] = acc[4 + k][r];
    }
  }
}

extern "C" void kernel_launch(void* const* d_in, const int* in_sizes, int n_in,
                              void* d_out, int out_size, void* d_ws, size_t ws_size,
                              hipStream_t stream) {
  const float* feats = (const float*)d_in[0];
  const float* msgs  = (const float*)d_in[1];
  WPtrs wp;
  for (int p = 0; p < 16; ++p) wp.w[p] = (const float*)d_in[2 + p];
  float* out = (float*)d_out;
  float* scratch = (float*)d_ws;                      // needs 2*NFRAG*64*4 = 176 KB

  // 1) Build pre-scaled, pre-masked B-fragment stream (deterministic every call).
  hipLaunchKernelGGL(prep_kernel, dim3(2 * NFRAG), dim3(64), 0, stream, wp, scratch);

  // 2) Main fused tensor-square kernel.
  const int nNodes = in_sizes[0] / DIM;
  const int nTiles = (nNodes + TILE - 1) / TILE;
  dim3 grid((nTiles + WPB - 1) / WPB);
  dim3 block(THREADS);
  const size_t shmem = (size_t)WPB * 2 * TILE * LDS_STRIDE * sizeof(float); // 55296 B
  hipLaunchKernelGGL(tsq_kernel, grid, block, shmem, stream, feats, msgs, out, scratch, nNodes);
}